// TGNModel_62362925138543
// MI455X (gfx1250) — compile-verified
//
#include <hip/hip_runtime.h>
#include <hip/hip_bf16.h>
#include <math.h>

typedef float v2f __attribute__((ext_vector_type(2)));
typedef float v8f __attribute__((ext_vector_type(8)));

#define HDIM 64
#define TEDIM 32
#define EADIM 32
#define DFEAT 128
#define FEAT 192      // 2H + EA + TE
#define G4 256        // 4H
#define ACLS 50
#define CHUNK 64      // events per chunk
#define ROWS 128      // 2 LSTM rows per event
#define RSTR 68       // padded LDS row stride: bank = (4*row + k) % 64, conflict-free b64
#define ASTR 196      // embed A-tile stride (196 % 64 == 4, same property)
#define ZSTR 68

__device__ __forceinline__ float sigmoidf_(float x) { return 1.0f / (1.0f + expf(-x)); }

__device__ __forceinline__ float agent_ld(const float* p) {
    return __hip_atomic_load(p, __ATOMIC_RELAXED, __HIP_MEMORY_SCOPE_AGENT);
}
__device__ __forceinline__ void agent_st(float* p, float v) {
    __hip_atomic_store(p, v, __ATOMIC_RELAXED, __HIP_MEMORY_SCOPE_AGENT);
}

// ---------------------------------------------------------------------------
extern "C" __global__ void tgn_zero(float* __restrict__ p, long n) {
    long i = (long)blockIdx.x * blockDim.x + threadIdx.x;
    long stride = (long)gridDim.x * blockDim.x;
    for (; i < n; i += stride) p[i] = 0.0f;
}

// ---------------------------------------------------------------------------
// Chunked TGN event scan. One workgroup (256 threads = 8 wave32).
// Weights pre-swizzled into LDS as [kb][q][lane] float2 so every WMMA
// B-fragment is one conflict-free ds_load_b64; tiles use 68-float row stride
// so A-fragments are conflict-free too. Per dependency level:
//   GEMM1 msg MLP [128,192]@[192,64], GEMM2 gates [128,128]@[128,256].
// ---------------------------------------------------------------------------
extern "C" __global__ __launch_bounds__(256) void tgn_scan(
    const int*   __restrict__ edge_index,   // [2,E]
    const float* __restrict__ edge_attr,    // [E,32]
    const float* __restrict__ edge_time,    // [E]
    const float* __restrict__ time_w, const float* __restrict__ time_b,
    const float* __restrict__ W_msg, const float* __restrict__ b_msg,
    const float* __restrict__ W_ih,  const float* __restrict__ W_hh,
    const float* __restrict__ b_lstm,
    float* __restrict__ hmem, float* __restrict__ cmem, int E)
{
    extern __shared__ float smem[];
    float* sWmS  = smem;                 // 48 kb * 4 q * 32 ln * 2 = 12288
    float* sWihS = sWmS  + 12288;        // 16 kb * 16 q * 32 * 2  = 16384
    float* sWhhS = sWihS + 16384;        // 16384
    float* sH    = sWhhS + 16384;        // 128*68 = 8704 (2e=hs, 2e+1=hd)
    float* sC    = sH  + ROWS*RSTR;      // 8704
    float* sET   = sC  + ROWS*RSTR;      // 64*68 = 4352 (per event: ea|te)
    float* sM    = sET + CHUNK*RSTR;     // 8704 (relu msg outputs)
    // total 75520 floats = 302,080 B

    __shared__ int nS[CHUNK], nD[CHUNK], lvl[CHUNK], vld[CHUNK];
    __shared__ float stw[TEDIM], stb[TEDIM];
    __shared__ int maxLvl, changed;

    const int t     = threadIdx.x;
    const int wave  = t >> 5;
    const int lane  = t & 31;
    const int lhalf = lane >> 4;
    const int l16   = lane & 15;

    if (t < TEDIM) { stw[t] = time_w[t]; stb[t] = time_b[t]; }

    // ---- one-time: swizzle weights into LDS ------------------------------
    // pair p = (kb*NT + q)*32 + ln ; src row kk = 4*kb + 2*(ln>>4) (+j)
    for (int p = t; p < 48*4*32; p += 256) {          // W_msg [192,64]
        int ln = p & 31, rest = p >> 5, q = rest & 3, kb = rest >> 2;
        int kk = 4*kb + ((ln >> 4) << 1);
        int n  = q*16 + (ln & 15);
        sWmS[2*p]   = W_msg[(long)kk*HDIM + n];
        sWmS[2*p+1] = W_msg[(long)(kk+1)*HDIM + n];
    }
    for (int p = t; p < 16*16*32; p += 256) {         // W_ih / W_hh [64,256]
        int ln = p & 31, rest = p >> 5, q = rest & 15, kb = rest >> 4;
        int kk = 4*kb + ((ln >> 4) << 1);
        int n  = q*16 + (ln & 15);
        sWihS[2*p]   = W_ih[(long)kk*G4 + n];
        sWihS[2*p+1] = W_ih[(long)(kk+1)*G4 + n];
        sWhhS[2*p]   = W_hh[(long)kk*G4 + n];
        sWhhS[2*p+1] = W_hh[(long)(kk+1)*G4 + n];
    }
    __syncthreads();

    for (int cb = 0; cb < E; cb += CHUNK) {
        // ---- chunk setup --------------------------------------------------
        if (t < CHUNK) {
            int ok = (cb + t) < E;
            vld[t] = ok;
            nS[t]  = ok ? edge_index[cb + t]     : -1;
            nD[t]  = ok ? edge_index[E + cb + t] : -1;
            lvl[t] = ok ? 0 : 0x40000000;
        }
        if (t == 0) maxLvl = 0;
        __syncthreads();

        for (int i = t; i < CHUNK*64; i += 256) {     // [ea | te] rows
            int ev = i >> 6, k = i & 63;
            float v = 0.0f;
            if (vld[ev]) {
                if (k < EADIM) v = edge_attr[(long)(cb+ev)*EADIM + k];
                else           v = cosf(edge_time[cb+ev] * stw[k-EADIM] + stb[k-EADIM]);
            }
            sET[ev*RSTR + k] = v;
        }
        if (t == 0 && cb + CHUNK < E) {
            __builtin_prefetch(edge_attr + (long)(cb+CHUNK)*EADIM, 0, 1);
            __builtin_prefetch(edge_index + cb + CHUNK, 0, 1);
        }

        // ---- intra-chunk dependency levels (Jacobi fixpoint) -------------
        while (true) {
            if (t == 0) changed = 0;
            __syncthreads();
            if (t < CHUNK && vld[t]) {
                int s = nS[t], d = nD[t], cur = lvl[t], nl = cur;
                for (int j = 0; j < t; ++j) {
                    if (!vld[j]) continue;
                    int sj = nS[j], dj = nD[j];
                    if (sj == s || sj == d || dj == s || dj == d) {
                        int cand = lvl[j] + 1;
                        nl = (cand > nl) ? cand : nl;
                    }
                }
                if (nl != cur) { lvl[t] = nl; atomicMax(&maxLvl, nl); changed = 1; }
            }
            __syncthreads();
            if (!changed) break;
        }
        const int ML = maxLvl;

        for (int L = 0; L <= ML; ++L) {
            // ---- gather active rows from L2-resident node state ----------
            for (int i = t; i < ROWS*HDIM; i += 256) {
                int row = i >> 6, k = i & 63;
                int ev = row >> 1;
                if (lvl[ev] == L) {
                    long node = (row & 1) ? nD[ev] : nS[ev];
                    sH[row*RSTR + k] = agent_ld(hmem + node*HDIM + k);
                    sC[row*RSTR + k] = agent_ld(cmem + node*HDIM + k);
                }
            }
            __syncthreads();

            const int m0   = wave*16 + l16;
            const int koff = lhalf * 2;

            // ---- GEMM1: M1 = relu(Feat @ W_msg + b)  ---------------------
            // Feat row 2e = [hd | hs | ea te], 2e+1 = [hs | hd | ea te]
            {
                const float* a0 = sH  + (m0 ^ 1) * RSTR;   // k 0..63
                const float* a1 = sH  + m0 * RSTR;         // k 64..127
                const float* a2 = sET + (m0 >> 1) * RSTR;  // k 128..191
                v8f acc[4] = {};
                #pragma unroll 4
                for (int k = 0; k < 64; k += 4) {
                    v2f a = *(const v2f*)(a0 + k + koff);
                    const float* wb = sWmS + (((k >> 2) * 4) * 32 + lane) * 2;
                    #pragma unroll
                    for (int q = 0; q < 4; ++q) {
                        v2f b = *(const v2f*)(wb + q * 64);
                        acc[q] = __builtin_amdgcn_wmma_f32_16x16x4_f32(
                                     false, a, false, b, (short)0, acc[q], false, false);
                    }
                }
                #pragma unroll 4
                for (int k = 0; k < 64; k += 4) {
                    v2f a = *(const v2f*)(a1 + k + koff);
                    const float* wb = sWmS + (((16 + (k >> 2)) * 4) * 32 + lane) * 2;
                    #pragma unroll
                    for (int q = 0; q < 4; ++q) {
                        v2f b = *(const v2f*)(wb + q * 64);
                        acc[q] = __builtin_amdgcn_wmma_f32_16x16x4_f32(
                                     false, a, false, b, (short)0, acc[q], false, false);
                    }
                }
                #pragma unroll 4
                for (int k = 0; k < 64; k += 4) {
                    v2f a = *(const v2f*)(a2 + k + koff);
                    const float* wb = sWmS + (((32 + (k >> 2)) * 4) * 32 + lane) * 2;
                    #pragma unroll
                    for (int q = 0; q < 4; ++q) {
                        v2f b = *(const v2f*)(wb + q * 64);
                        acc[q] = __builtin_amdgcn_wmma_f32_16x16x4_f32(
                                     false, a, false, b, (short)0, acc[q], false, false);
                    }
                }
                #pragma unroll
                for (int q = 0; q < 4; ++q) {
                    int n = q*16 + l16;
                    float bm = b_msg[n];
                    #pragma unroll
                    for (int v = 0; v < 8; ++v) {
                        int m = wave*16 + lhalf*8 + v;
                        sM[m*RSTR + n] = fmaxf(acc[q][v] + bm, 0.0f);
                    }
                }
            }
            // no barrier: each wave reads back only its own sM/sH/sC rows

            // ---- GEMM2 + gates, one N-tile group (i,f,g,o) at a time -----
            #pragma unroll
            for (int tt = 0; tt < 4; ++tt) {
                v8f acc[4] = {};   // q = tt, tt+4, tt+8, tt+12
                #pragma unroll 4
                for (int k = 0; k < 64; k += 4) {       // M1 @ W_ih
                    v2f a = *(const v2f*)(sM + m0*RSTR + k + koff);
                    const float* wb = sWihS + (((k >> 2) * 16) * 32 + lane) * 2;
                    #pragma unroll
                    for (int j = 0; j < 4; ++j) {
                        v2f b = *(const v2f*)(wb + (tt + 4*j) * 64);
                        acc[j] = __builtin_amdgcn_wmma_f32_16x16x4_f32(
                                     false, a, false, b, (short)0, acc[j], false, false);
                    }
                }
                #pragma unroll 4
                for (int k = 0; k < 64; k += 4) {       // H @ W_hh
                    v2f a = *(const v2f*)(sH + m0*RSTR + k + koff);
                    const float* wb = sWhhS + (((k >> 2) * 16) * 32 + lane) * 2;
                    #pragma unroll
                    for (int j = 0; j < 4; ++j) {
                        v2f b = *(const v2f*)(wb + (tt + 4*j) * 64);
                        acc[j] = __builtin_amdgcn_wmma_f32_16x16x4_f32(
                                     false, a, false, b, (short)0, acc[j], false, false);
                    }
                }
                int n = tt*16 + l16;
                float bi = b_lstm[n];
                float bf = b_lstm[HDIM   + n];
                float bg = b_lstm[2*HDIM + n];
                float bo = b_lstm[3*HDIM + n];
                #pragma unroll
                for (int v = 0; v < 8; ++v) {
                    int m  = wave*16 + lhalf*8 + v;
                    int ev = m >> 1;
                    float gi = acc[0][v] + bi;
                    float gf = acc[1][v] + bf;
                    float gg = acc[2][v] + bg;
                    float go = acc[3][v] + bo;
                    float cold = sC[m*RSTR + n];
                    float cn = sigmoidf_(gf)*cold + sigmoidf_(gi)*tanhf(gg);
                    float hn = sigmoidf_(go)*tanhf(cn);
                    if (lvl[ev] == L) {
                        long node = (m & 1) ? nD[ev] : nS[ev];
                        if (nS[ev] != nD[ev] || (m & 1)) {      // dup: dst wins
                            agent_st(hmem + node*HDIM + n, hn);
                            agent_st(cmem + node*HDIM + n, cn);
                        }
                    }
                }
            }
            __threadfence();
            __syncthreads();
        } // levels
    } // chunks
}

// ---------------------------------------------------------------------------
// Embedding + classifier via f32 WMMA, only for edge-dst rows.
// ---------------------------------------------------------------------------
extern "C" __global__ __launch_bounds__(128) void tgn_embed(
    const int*   __restrict__ edge_dst,
    const float* __restrict__ x,
    const float* __restrict__ hmem,
    const float* __restrict__ W_emb, const float* __restrict__ b_emb,
    const float* __restrict__ W_cls, const float* __restrict__ b_cls,
    float* __restrict__ out, int E)
{
    __shared__ float Ald[16*ASTR];   // gathered [h || x] tile (padded stride)
    __shared__ float Zld[16*ZSTR];   // embedding tile

    const int tid   = threadIdx.x;
    const int wave  = tid >> 5;
    const int lane  = tid & 31;
    const int ebase = blockIdx.x * 16;

    for (int i = tid; i < 16*FEAT; i += 128) {
        int row = i / FEAT, col = i - row*FEAT;
        int d = edge_dst[ebase + row];
        Ald[row*ASTR + col] = (col < HDIM) ? hmem[(long)d*HDIM + col]
                                           : x[(long)d*DFEAT + (col - HDIM)];
    }
    __syncthreads();

    const int arow = lane & 15;
    const int koff = (lane >> 4) * 2;
    const int ncol = (lane & 15) + wave * 16;

    v8f acc = {};
    #pragma unroll 4
    for (int k = 0; k < FEAT; k += 4) {
        v2f a = *(const v2f*)(Ald + arow*ASTR + k + koff);
        v2f b;
        b.x = W_emb[(long)(k + koff)    *HDIM + ncol];
        b.y = W_emb[(long)(k + koff + 1)*HDIM + ncol];
        acc = __builtin_amdgcn_wmma_f32_16x16x4_f32(false, a, false, b,
                                                    (short)0, acc, false, false);
    }
    {
        float be = b_emb[ncol];
        int mb = (lane >> 4) * 8;
        #pragma unroll
        for (int v = 0; v < 8; ++v)
            Zld[(mb + v)*ZSTR + ncol] = acc[v] + be;
    }
    __syncthreads();

    v8f acc2 = {};
    const bool valid = (ncol < ACLS);
    #pragma unroll 4
    for (int k = 0; k < HDIM; k += 4) {
        v2f a = *(const v2f*)(Zld + arow*ZSTR + k + koff);
        v2f b;
        b.x = valid ? W_cls[(long)(k + koff)    *ACLS + ncol] : 0.0f;
        b.y = valid ? W_cls[(long)(k + koff + 1)*ACLS + ncol] : 0.0f;
        acc2 = __builtin_amdgcn_wmma_f32_16x16x4_f32(false, a, false, b,
                                                     (short)0, acc2, false, false);
    }
    if (valid) {
        float bc = b_cls[ncol];
        int mb = (lane >> 4) * 8;
        #pragma unroll
        for (int v = 0; v < 8; ++v) {
            int e = ebase + mb + v;
            out[(long)e*ACLS + ncol] = acc2[v] + bc;
        }
    }
}

// ---------------------------------------------------------------------------
extern "C" void kernel_launch(void* const* d_in, const int* in_sizes, int n_in,
                              void* d_out, int out_size, void* d_ws, size_t ws_size,
                              hipStream_t stream) {
    const float* x          = (const float*)d_in[0];
    const int*   edge_index = (const int*)  d_in[1];
    const float* edge_attr  = (const float*)d_in[2];
    const float* edge_time  = (const float*)d_in[3];
    const float* time_w     = (const float*)d_in[4];
    const float* time_b     = (const float*)d_in[5];
    const float* W_msg      = (const float*)d_in[6];
    const float* b_msg      = (const float*)d_in[7];
    const float* W_ih       = (const float*)d_in[8];
    const float* W_hh       = (const float*)d_in[9];
    const float* b_lstm     = (const float*)d_in[10];
    const float* W_emb      = (const float*)d_in[11];
    const float* b_emb      = (const float*)d_in[12];
    const float* W_cls      = (const float*)d_in[13];
    const float* b_cls      = (const float*)d_in[14];
    float* out = (float*)d_out;

    const int N = in_sizes[0] / DFEAT;
    const int E = in_sizes[3];

    float* hmem = (float*)d_ws;            // [N, 64]
    float* cmem = hmem + (long)N * HDIM;   // [N, 64]

    // 1) zero node memory + cell state
    tgn_zero<<<2048, 256, 0, stream>>>(hmem, 2L * N * HDIM);

    // 2) chunk-parallel event scan, weights swizzled in LDS, WMMA math
    const size_t scan_lds = (size_t)(12288 + 2*16384 + 3*ROWS*RSTR + CHUNK*RSTR)
                            * sizeof(float);   // 302,080 B
    hipFuncSetAttribute((const void*)tgn_scan,
                        hipFuncAttributeMaxDynamicSharedMemorySize, (int)scan_lds);
    tgn_scan<<<1, 256, scan_lds, stream>>>(edge_index, edge_attr, edge_time,
                                           time_w, time_b, W_msg, b_msg,
                                           W_ih, W_hh, b_lstm, hmem, cmem, E);

    // 3) gathered embedding + classifier (WMMA)
    const int* edge_dst = edge_index + E;
    tgn_embed<<<(E + 15)/16, 128, 0, stream>>>(edge_dst, x, hmem, W_emb, b_emb,
                                               W_cls, b_cls, out, E);
}